// FeatureLoss_13907104104609
// MI455X (gfx1250) — compile-verified
//
#include <hip/hip_runtime.h>

#define B_ 8
#define C_ 128
#define H_ 192
#define W_ 192
#define HW_ (H_ * W_)
#define K_ 1024
#define REG_ 7
#define RC_ 49        // 7*7
#define MSTRIDE 64    // mean vectors padded 49 -> 64 (16 WMMA K=4 steps)
#define NT 64         // K_/16 tiles per dimension

typedef float v2f __attribute__((ext_vector_type(2)));
typedef float v8f __attribute__((ext_vector_type(8)));

// ---------------------------------------------------------------------------
// Kernel 1: peak detection + deterministic ordered compaction (one block/img)
// ---------------------------------------------------------------------------
__global__ void peaks_kernel(const float* __restrict__ heat,
                             int* __restrict__ counts,
                             int* __restrict__ peaks) {
  __shared__ int s[1024];
  const int b = blockIdx.x;
  const int tid = threadIdx.x;
  const float* hb = heat + (size_t)b * HW_;
  int base = 0;
  for (int chunk = 0; chunk < HW_ / 1024; ++chunk) {
    const int pix = chunk * 1024 + tid;
    const int r = pix / W_, c = pix % W_;
    const float h = hb[pix];
    int flag = 0;
    if (h > 0.0f) {
      float m = h;
#pragma unroll
      for (int dr = -1; dr <= 1; ++dr)
#pragma unroll
        for (int dc = -1; dc <= 1; ++dc) {
          const int rr = r + dr, cc = c + dc;
          if (rr >= 0 && rr < H_ && cc >= 0 && cc < W_)
            m = fmaxf(m, hb[rr * W_ + cc]);
        }
      flag = (h >= m) ? 1 : 0;  // equals 3x3 max and positive -> peak
    }
    s[tid] = flag;
    __syncthreads();
    for (int off = 1; off < 1024; off <<= 1) {
      const int v = (tid >= off) ? s[tid - off] : 0;
      __syncthreads();
      s[tid] += v;
      __syncthreads();
    }
    const int incl = s[tid];
    const int total = s[1023];
    if (flag) {
      const int pos = base + incl - 1;
      if (pos < K_) peaks[b * K_ + pos] = pix;
    }
    base += total;
    __syncthreads();
  }
  if (tid == 0) counts[b] = (base > K_) ? K_ : base;
}

// ---------------------------------------------------------------------------
// Kernel 2: per-peak region means (zero-padded to MSTRIDE), sq-norms, intra
// loss term. One 64-thread block per (k, b). Invalid slots zeroed.
// ---------------------------------------------------------------------------
__global__ void region_kernel(const float* __restrict__ feat,
                              const int* __restrict__ counts,
                              const int* __restrict__ peaks,
                              float* __restrict__ means,
                              float* __restrict__ sqv,
                              float* __restrict__ perreg) {
  const int k = blockIdx.x, b = blockIdx.y;
  const int tid = threadIdx.x;  // 0..63
  int n = counts[b];
  if (n > K_) n = K_;
  float* mrow = means + (size_t)(b * K_ + k) * MSTRIDE;
  if (k >= n) {
    mrow[tid] = 0.0f;
    if (tid == 0) { sqv[b * K_ + k] = 0.0f; perreg[b * K_ + k] = 0.0f; }
    return;
  }
  const int pix = peaks[b * K_ + k];
  const int pr = pix / W_, pc = pix % W_;
  const int row0 = pc - 3, col0 = pr - 3;  // reference's intentional x/y swap
  const float* fb = feat + (size_t)b * C_ * HW_;

  __shared__ float smean[RC_];
  if (tid < RC_) {
    const int i = tid / REG_, j = tid % REG_;
    const int rr = row0 + i, cc = col0 + j;
    float s = 0.0f;
    if (rr >= 0 && rr < H_ && cc >= 0 && cc < W_) {
      const float* p = fb + rr * W_ + cc;
      for (int ch = 0; ch < C_; ++ch) s += p[(size_t)ch * HW_];
    }
    const float mn = s * (1.0f / C_);
    smean[tid] = mn;
    mrow[tid] = mn;
  } else {
    mrow[tid] = 0.0f;  // pad entries 49..63
  }
  __syncthreads();
  if (tid == 0) {
    float s2 = 0.0f;
    for (int p = 0; p < RC_; ++p) s2 += smean[p] * smean[p];
    sqv[b * K_ + k] = s2;
  }
  float dsum = 0.0f;
  for (int ch = tid; ch < C_; ch += 64) {
    const float* p = fb + (size_t)ch * HW_;
    float d2 = 0.0f;
    for (int pos = 0; pos < RC_; ++pos) {
      const int i = pos / REG_, j = pos % REG_;
      const int rr = row0 + i, cc = col0 + j;
      const float v = (rr >= 0 && rr < H_ && cc >= 0 && cc < W_)
                          ? p[rr * W_ + cc] : 0.0f;
      const float df = v - smean[pos];
      d2 += df * df;
    }
    dsum += (d2 > 0.0f) ? sqrtf(d2) : 0.0f;
  }
  __shared__ float red[64];
  red[tid] = dsum;
  __syncthreads();
  for (int off = 32; off; off >>= 1) {
    if (tid < off) red[tid] += red[tid + off];
    __syncthreads();
  }
  if (tid == 0) perreg[b * K_ + k] = red[0] * (1.0f / C_);
}

// ---------------------------------------------------------------------------
// Kernel 3: pairwise hinge loss via fp32 WMMA Gram tiles. One wave per 16x16
// tile (EXEC all ones, uniform branches only). Deterministic per-tile output.
// ---------------------------------------------------------------------------
__global__ void pair_kernel(const int* __restrict__ counts,
                            const float* __restrict__ means,
                            const float* __restrict__ sqv,
                            float* __restrict__ tiles) {
  const int ti = blockIdx.x, tj = blockIdx.y, b = blockIdx.z;
  const int lane = threadIdx.x;  // 0..31, wave32
  int n = counts[b];
  if (n > K_) n = K_;
  float* tout = tiles + ((size_t)b * NT + ti) * NT + tj;
  if (tj < ti || ti * 16 >= n || tj * 16 >= n) {  // uniform across the wave
    if (lane == 0) *tout = 0.0f;
    return;
  }
  const int I = ti * 16, J = tj * 16;
  const int half = lane >> 4, lr = lane & 15;
  const float* mb = means + (size_t)b * K_ * MSTRIDE;
  // A 16x4 f32 layout: lane L -> M = L&15, K = (L>>4)*2 + {0,1} (per VGPR)
  // B 4x16 f32 layout: lane L -> N = L&15, K = (L>>4)*2 + {0,1} -> same
  // addressing pattern since B = mf[J..J+15]^T.
  const float* ar = mb + (size_t)(I + lr) * MSTRIDE + half * 2;
  const float* br = mb + (size_t)(J + lr) * MSTRIDE + half * 2;
  v8f acc = {0.f, 0.f, 0.f, 0.f, 0.f, 0.f, 0.f, 0.f};
#pragma unroll
  for (int kk = 0; kk < 16; ++kk) {
    v2f a; a.x = ar[kk * 4]; a.y = ar[kk * 4 + 1];
    v2f bb; bb.x = br[kk * 4]; bb.y = br[kk * 4 + 1];
    acc = __builtin_amdgcn_wmma_f32_16x16x4_f32(
        /*neg_a=*/false, a, /*neg_b=*/false, bb,
        /*c_mod=*/(short)0, acc, /*reuse_a=*/false, /*reuse_b=*/false);
  }
  // C/D layout: VGPR v, lane L -> M = v + 8*(L>>4), N = L&15
  const float* sqb = sqv + (size_t)b * K_;
  const int gj = J + lr;
  const float sqj = sqb[gj];
  float sum = 0.0f;
#pragma unroll
  for (int v = 0; v < 8; ++v) {
    const int gi = I + v + 8 * half;
    if (gi < gj && gj < n) {  // gi<gj<n implies gi<n
      float d2 = sqb[gi] + sqj - 2.0f * acc[v];
      d2 = fmaxf(d2, 0.0f);
      const float d = sqrtf(d2);
      sum += fmaxf(0.0f, 1.0f - d);
    }
  }
#pragma unroll
  for (int off = 16; off; off >>= 1) sum += __shfl_xor(sum, off, 32);
  if (lane == 0) *tout = sum;
}

// ---------------------------------------------------------------------------
// Kernel 4: fixed-order final reduction -> scalar loss
// ---------------------------------------------------------------------------
__global__ void final_kernel(const int* __restrict__ counts,
                             const float* __restrict__ perreg,
                             const float* __restrict__ tiles,
                             float* __restrict__ out) {
  __shared__ float red[256];
  const int tid = threadIdx.x;
  float acc_in = 0.0f, acc_out = 0.0f;  // meaningful on tid 0
  for (int b = 0; b < B_; ++b) {
    int n = counts[b];
    if (n > K_) n = K_;
    float s = 0.0f;
    for (int k = tid; k < K_; k += 256) s += perreg[b * K_ + k];
    red[tid] = s;
    __syncthreads();
    for (int off = 128; off; off >>= 1) {
      if (tid < off) red[tid] += red[tid + off];
      __syncthreads();
    }
    const float sum_in = red[0];
    __syncthreads();
    float s2 = 0.0f;
    for (int t = tid; t < NT * NT; t += 256) s2 += tiles[b * NT * NT + t];
    red[tid] = s2;
    __syncthreads();
    for (int off = 128; off; off >>= 1) {
      if (tid < off) red[tid] += red[tid + off];
      __syncthreads();
    }
    const float sum_out = red[0];
    __syncthreads();
    if (tid == 0) {
      const float fn = (float)n;
      if (n > 0) acc_in += sum_in / fn;
      const float npairs = fn * (fn - 1.0f) * 0.5f;
      if (npairs > 0.0f) acc_out += sum_out / npairs;
    }
  }
  if (tid == 0)
    out[0] = 0.5f * (acc_in / (float)B_) + 1.0f * (acc_out / (float)B_);
}

// ---------------------------------------------------------------------------
extern "C" void kernel_launch(void* const* d_in, const int* in_sizes, int n_in,
                              void* d_out, int out_size, void* d_ws,
                              size_t ws_size, hipStream_t stream) {
  (void)in_sizes; (void)n_in; (void)out_size; (void)ws_size;
  const float* feat = (const float*)d_in[0];  // [8,128,192,192] f32
  const float* heat = (const float*)d_in[1];  // [8,1,192,192] f32
  float* out = (float*)d_out;

  char* ws = (char*)d_ws;
  constexpr size_t OFF_COUNT = 0;                                    // 8 ints
  constexpr size_t OFF_PEAKS = 64;                                   // 8K ints
  constexpr size_t OFF_MEANS = OFF_PEAKS + (size_t)B_ * K_ * 4;      // 2 MB f32
  constexpr size_t OFF_SQ    = OFF_MEANS + (size_t)B_ * K_ * MSTRIDE * 4;
  constexpr size_t OFF_PR    = OFF_SQ + (size_t)B_ * K_ * 4;
  constexpr size_t OFF_TILES = OFF_PR + (size_t)B_ * K_ * 4;

  int* counts   = (int*)(ws + OFF_COUNT);
  int* peaks    = (int*)(ws + OFF_PEAKS);
  float* means  = (float*)(ws + OFF_MEANS);
  float* sqv    = (float*)(ws + OFF_SQ);
  float* perreg = (float*)(ws + OFF_PR);
  float* tiles  = (float*)(ws + OFF_TILES);

  peaks_kernel<<<B_, 1024, 0, stream>>>(heat, counts, peaks);
  region_kernel<<<dim3(K_, B_), 64, 0, stream>>>(feat, counts, peaks, means,
                                                 sqv, perreg);
  pair_kernel<<<dim3(NT, NT, B_), 32, 0, stream>>>(counts, means, sqv, tiles);
  final_kernel<<<1, 256, 0, stream>>>(counts, perreg, tiles, out);
}